// OriginalCANet_48352741818659
// MI455X (gfx1250) — compile-verified
//
#include <hip/hip_runtime.h>

// ---------------------------------------------------------------------------
// 3x3 'SAME' conv (1->1 ch) + bias + threshold activation, 16x2048x2048 fp32.
// Memory-bound (AI ~2.2 FLOP/B, ~23us floor at 23.3 TB/s) -> optimize the
// data path with CDNA5 features:
//   * TDM tensor_load_to_lds DMAs each (256 x 18) input tile global->LDS,
//     using D# row padding (256 dwords + 8 dword gap) so LDS rows become
//     [4 halo][256 main][4 halo] = 264 floats, contiguous.
//   * Exactly ONE TDM descriptor per tile: issue gated by a wave-uniform
//     scalar branch (readfirstlane) since TDM ignores EXEC.
//   * Halo columns / zero-padding rows via global_load_async_to_lds_b128
//     (ASYNCcnt) and DS zero stores, disjoint from TDM destinations.
//   * Double-buffered pipeline: prefetch tile t+1 (TDM) while computing t;
//     s_wait_tensorcnt 1 keeps one DMA in flight.
//   * Non-temporal b128 stores for the streamed output (don't pollute L2;
//     L2 then serves vertical-halo re-reads across y-adjacent tiles).
// Activation identity: C*exp(-0.5*(s-2.6)^2) >= C*exp(-1/8)
//                  <=> |s-2.6| <= 0.5 <=> 2.1 <= s <= 3.1.
// ---------------------------------------------------------------------------

typedef float        v4f   __attribute__((ext_vector_type(4)));
typedef unsigned int u32x4 __attribute__((ext_vector_type(4)));
typedef unsigned int u32x8 __attribute__((ext_vector_type(8)));

#define W_IMG 2048
#define H_IMG 2048
#define TILE_W 256                      // tile width in pixels (1 TDM row)
#define TILE_H 16                       // output rows per tile
#define CHUNK_ROWS 128                  // rows per workgroup (strip)
#define NT (CHUNK_ROWS / TILE_H)        // 8 tiles per workgroup
#define LDS_ROWS (TILE_H + 2)           // 18 staged rows (incl. vert halo)
#define ROW_FLOATS 264                  // 4 halo + 256 main + 4 halo
#define BUF_FLOATS (LDS_ROWS * ROW_FLOATS)  // 4752 floats per buffer

// Issue one TDM descriptor: load (TILE_W x rows) fp32 tile into LDS with an
// 8-dword pad after every 256 dwords (-> 264-float LDS row stride).
__device__ static inline void tdm_load_tile(const float* src, int x0, int y0t,
                                            unsigned bufByteBase)
{
    const int gy_start = y0t - 1;
    const int top = (gy_start < 0) ? 1 : 0;                 // clamp top halo
    const int bot = ((y0t + TILE_H) >= H_IMG) ? 1 : 0;      // clamp bottom
    const unsigned rows = (unsigned)(LDS_ROWS - top - bot);
    const unsigned long long ga =
        (unsigned long long)(src + (size_t)(gy_start + top) * W_IMG + (size_t)x0);

    u32x4 g0;
    g0.x = 1u;                                              // count=1 (valid)
    g0.y = bufByteBase + (unsigned)((top * ROW_FLOATS + 4) * 4); // lds_addr
    g0.z = (unsigned)ga;                                    // global_addr lo
    g0.w = (unsigned)((ga >> 32) & 0x1FFFFFFull) | (2u << 30);   // hi | type=2

    u32x8 g1;
    g1[0] = (2u << 16)      // data_size = 4 bytes
          | (1u << 20)      // pad_enable
          | (7u << 22)      // pad_interval code 7 -> every 256 dwords
          | (7u << 25);     // pad_amount  code 7 -> 8 dwords (32B halo gap)
    g1[1] = ((unsigned)TILE_W & 0xFFFFu) << 16;   // tensor_dim0[15:0]
    g1[2] = (rows & 0xFFFFu) << 16;               // dim0 hi=0 | tensor_dim1 lo
    g1[3] = ((unsigned)TILE_W) << 16;             // dim1 hi=0 | tile_dim0
    g1[4] = rows & 0xFFFFu;                       // tile_dim1 | tile_dim2=0
    g1[5] = (unsigned)W_IMG;                      // tensor_dim0_stride lo32
    g1[6] = 0u;                                   // stride hi | dim1_stride lo
    g1[7] = 0u;                                   // dim1_stride hi
    asm volatile("tensor_load_to_lds %0, %1" :: "s"(g0), "s"(g1) : "memory");
}

__global__ __launch_bounds__(256) void canet_conv3x3_act(
    const float* __restrict__ x, const float* __restrict__ wgt,
    const float* __restrict__ bias, float* __restrict__ out)
{
    __shared__ float tile[2 * BUF_FLOATS];          // 38,016 bytes, ping-pong

    const int tid = (int)threadIdx.x;
    const int x0  = (int)blockIdx.x * TILE_W;
    const int yc  = (int)blockIdx.y * CHUNK_ROWS;
    const int b   = (int)blockIdx.z;

    const float* src = x   + (size_t)b * ((size_t)W_IMG * H_IMG);
    float*       dst = out + (size_t)b * ((size_t)W_IMG * H_IMG);

    // Hardware LDS byte offset of tile[0] = low 32 bits of generic pointer.
    const unsigned ldsBase = (unsigned)(unsigned long long)(void*)&tile[0];

    // Wave-uniform scalar predicate: true only for wave 0 (first lane tid==0).
    // TDM ignores EXEC, so the guard must be a *scalar* branch, not an EXEC
    // mask, or every wave would issue a duplicate DMA descriptor.
    const bool tdmWave = (__builtin_amdgcn_readfirstlane(tid) == 0);

    // Broadcast weights / bias (uniform -> SGPRs via s_load).
    const float w00 = wgt[0], w01 = wgt[1], w02 = wgt[2];
    const float w10 = wgt[3], w11 = wgt[4], w12 = wgt[5];
    const float w20 = wgt[6], w21 = wgt[7], w22 = wgt[8];
    const float bs  = bias[0];

    const int c4 = tid & 63;        // float4 column within tile
    const int rb = tid >> 6;        // row-group 0..3 (rows rb, rb+4, rb+8, rb+12)

    // Prefetch tile 0 into buffer 0 (wave 0 drives the TDM).
    if (tdmWave) tdm_load_tile(src, x0, yc, ldsBase);

    for (int t = 0; t < NT; ++t) {
        const int y0t = yc + t * TILE_H;
        const unsigned bufF = (unsigned)((t & 1) * BUF_FLOATS);   // float idx

        // ---- halo columns + clamped-row zero fill for buf[t&1] ----------
        // (all destinations disjoint from the TDM main region)
        const int gy_start = y0t - 1;
        const int top = (gy_start < 0) ? 1 : 0;
        const int bot = ((y0t + TILE_H) >= H_IMG) ? 1 : 0;
        if (tid < 2 * LDS_ROWS) {
            const int side = tid & 1;               // 0 = left, 1 = right
            const int k    = tid >> 1;              // staged row 0..17
            const int gy   = gy_start + k;
            if (gy >= 0 && gy < H_IMG) {            // OOB rows zeroed below
                const int gx = side ? (x0 + TILE_W) : (x0 - 4);
                const unsigned fOff = bufF + (unsigned)(k * ROW_FLOATS + (side ? 260 : 0));
                if (gx >= 0 && gx + 4 <= W_IMG) {
                    const unsigned ldsOff = ldsBase + fOff * 4u;
                    const unsigned long long ga =
                        (unsigned long long)(src + (size_t)gy * W_IMG + (size_t)gx);
                    asm volatile("global_load_async_to_lds_b128 %0, %1, off"
                                 :: "v"(ldsOff), "v"(ga) : "memory");
                } else {
                    *(v4f*)&tile[fOff] = (v4f)(0.0f);   // image x-edge padding
                }
            }
        }
        if (top && tid < ROW_FLOATS / 4)            // missing top halo row
            *(v4f*)&tile[bufF + 4 * tid] = (v4f)(0.0f);
        if (bot && tid < ROW_FLOATS / 4)            // missing bottom halo row
            *(v4f*)&tile[bufF + (LDS_ROWS - 1) * ROW_FLOATS + 4 * tid] = (v4f)(0.0f);

        // ---- prefetch next tile; wait for current tile's DMA ------------
        if (tdmWave) {
            if (t + 1 < NT) {
                tdm_load_tile(src, x0, y0t + TILE_H,
                              ldsBase + (unsigned)((((t + 1) & 1) * BUF_FLOATS) * 4));
                __builtin_amdgcn_s_wait_tensorcnt(1);   // tile t done, t+1 in flight
            } else {
                __builtin_amdgcn_s_wait_tensorcnt(0);
            }
        }
        asm volatile("s_wait_asynccnt 0x0" ::: "memory");  // my wave's halo loads
        __syncthreads();                                   // all waves + DS stores

        // ---- compute 16 output rows of this tile ------------------------
        #pragma unroll
        for (int i = 0; i < 4; ++i) {
            const int r = rb + 4 * i;                      // output row in tile
            const float* r0 = &tile[bufF + (r + 0) * ROW_FLOATS + 3 + 4 * c4];
            const float* r1 = &tile[bufF + (r + 1) * ROW_FLOATS + 3 + 4 * c4];
            const float* r2 = &tile[bufF + (r + 2) * ROW_FLOATS + 3 + 4 * c4];
            v4f o;
            #pragma unroll
            for (int j = 0; j < 4; ++j) {
                float s = bs;
                s = fmaf(w00, r0[j + 0], s);
                s = fmaf(w01, r0[j + 1], s);
                s = fmaf(w02, r0[j + 2], s);
                s = fmaf(w10, r1[j + 0], s);
                s = fmaf(w11, r1[j + 1], s);
                s = fmaf(w12, r1[j + 2], s);
                s = fmaf(w20, r2[j + 0], s);
                s = fmaf(w21, r2[j + 1], s);
                s = fmaf(w22, r2[j + 2], s);
                o[j] = (s >= 2.1f && s <= 3.1f) ? 1.0f : 0.0f;
            }
            v4f* op = (v4f*)(dst + (size_t)(y0t + r) * W_IMG
                                 + (size_t)(x0 + 4 * c4));
            __builtin_nontemporal_store(o, op);            // streaming output
        }
        __syncthreads();    // buf[t&1] fully consumed before it is re-DMA'd
    }
}

extern "C" void kernel_launch(void* const* d_in, const int* in_sizes, int n_in,
                              void* d_out, int out_size, void* d_ws, size_t ws_size,
                              hipStream_t stream) {
    (void)n_in; (void)out_size; (void)d_ws; (void)ws_size;
    const float* x    = (const float*)d_in[0];   // (B,1,2048,2048) fp32
    const float* wgt  = (const float*)d_in[1];   // (1,1,3,3) fp32
    const float* bias = (const float*)d_in[2];   // (1,) fp32
    float* out = (float*)d_out;

    const int batch = in_sizes[0] / (W_IMG * H_IMG);          // = 16
    dim3 grid(W_IMG / TILE_W, H_IMG / CHUNK_ROWS, batch);     // (8, 16, 16)
    canet_conv3x3_act<<<grid, dim3(256), 0, stream>>>(x, wgt, bias, out);
}